// Diag_29025388986544
// MI455X (gfx1250) — compile-verified
//
#include <hip/hip_runtime.h>
#include <cstdint>
#include <cstddef>

// out = (x_real * exp(betas_real), x_imag * exp(betas_imag)); pure bandwidth op.
// ~537 MB HBM traffic -> ~23 us floor @ 23.3 TB/s. No matmul => no WMMA; the
// CDNA5 feature that matters here is the async global->LDS copy engine
// (ASYNCcnt) + 128-bit accesses + non-temporal streaming stores.

typedef float v4f __attribute__((ext_vector_type(4)));
typedef int   v4i __attribute__((vector_size(4 * sizeof(int))));  // b128 payload type

#define CHUNK     1024   // floats of the h*w dimension per workgroup
#define BC_PER_WG    8   // (b*c) slices handled per workgroup (scale regs reused 8x)
#define THREADS    256   // 8 wave32s

#if defined(__HIP_DEVICE_COMPILE__) && defined(__gfx1250__) && \
    __has_builtin(__builtin_amdgcn_global_load_async_to_lds_b128) && \
    __has_builtin(__builtin_amdgcn_s_wait_asynccnt)
#define USE_ASYNC 1
#else
#define USE_ASYNC 0
#endif

__global__ __launch_bounds__(THREADS) void diag_scale_kernel(
    const float* __restrict__ xr, const float* __restrict__ xi,
    const float* __restrict__ br, const float* __restrict__ bi,
    float* __restrict__ out, long long hw, long long n_half)
{
    const int       tid     = threadIdx.x;
    const long long hw_base = (long long)blockIdx.x * CHUNK;
    const int       bc0     = blockIdx.y * BC_PER_WG;
    const int       lane_f  = tid * 4;               // float offset inside chunk

    // Per-thread scales live in registers across all BC_PER_WG reuses.
    v4f sr = *(const v4f*)(br + hw_base + lane_f);
    v4f si = *(const v4f*)(bi + hw_base + lane_f);
    sr.x = expf(sr.x); sr.y = expf(sr.y); sr.z = expf(sr.z); sr.w = expf(sr.w);
    si.x = expf(si.x); si.y = expf(si.y); si.z = expf(si.z); si.w = expf(si.w);

#if USE_ASYNC
    // [2 buffers][2 components][CHUNK floats] = 16 KB of the 320 KB/WGP LDS
    __shared__ float lds[2 * 2 * CHUNK];

    auto issue = [&](int j) {
        const long long g     = (long long)(bc0 + j) * hw + hw_base + lane_f;
        const int       lbase = (j & 1) * (2 * CHUNK) + lane_f;
        __builtin_amdgcn_global_load_async_to_lds_b128(
            (__attribute__((address_space(1))) v4i*)(xr + g),
            (__attribute__((address_space(3))) v4i*)(lds + lbase), 0, 0);
        __builtin_amdgcn_global_load_async_to_lds_b128(
            (__attribute__((address_space(1))) v4i*)(xi + g),
            (__attribute__((address_space(3))) v4i*)(lds + lbase + CHUNK), 0, 0);
    };

    issue(0);
    #pragma unroll
    for (int j = 0; j < BC_PER_WG; ++j) {
        if (j + 1 < BC_PER_WG) {
            issue(j + 1);                            // prefetch next slice
            __builtin_amdgcn_s_wait_asynccnt(2);     // iter j's 2 copies done (in-order)
        } else {
            __builtin_amdgcn_s_wait_asynccnt(0);
        }
        asm volatile("" ::: "memory");               // compiler ordering fence

        const int lbase = (j & 1) * (2 * CHUNK) + lane_f;
        // Each lane reads exactly the LDS bytes its own async op wrote -> no barrier.
        v4f vr = *(const v4f*)(lds + lbase);
        v4f vi = *(const v4f*)(lds + lbase + CHUNK);
        vr *= sr;
        vi *= si;

        const long long g = (long long)(bc0 + j) * hw + hw_base + lane_f;
        // Streaming stores: outputs never re-read; keep L2 for the betas.
        __builtin_nontemporal_store(vr, (v4f*)(out + g));
        __builtin_nontemporal_store(vi, (v4f*)(out + n_half + g));
    }
#else
    // Fallback: direct 128-bit loads (still bandwidth-optimal, no async engine).
    #pragma unroll
    for (int j = 0; j < BC_PER_WG; ++j) {
        const long long g = (long long)(bc0 + j) * hw + hw_base + lane_f;
        v4f vr = __builtin_nontemporal_load((const v4f*)(xr + g));
        v4f vi = __builtin_nontemporal_load((const v4f*)(xi + g));
        vr *= sr;
        vi *= si;
        __builtin_nontemporal_store(vr, (v4f*)(out + g));
        __builtin_nontemporal_store(vi, (v4f*)(out + n_half + g));
    }
#endif
}

extern "C" void kernel_launch(void* const* d_in, const int* in_sizes, int n_in,
                              void* d_out, int out_size, void* d_ws, size_t ws_size,
                              hipStream_t stream) {
    const float* xr = (const float*)d_in[0];
    const float* xi = (const float*)d_in[1];
    const float* br = (const float*)d_in[2];
    const float* bi = (const float*)d_in[3];
    float* out = (float*)d_out;

    const long long hw = in_sizes[2];                  // h*w = 65536
    const long long bc = (long long)in_sizes[0] / hw;  // b*c = 512
    const long long n_half = bc * hw;                  // elements per component

    dim3 grid((unsigned)(hw / CHUNK), (unsigned)(bc / BC_PER_WG));  // (64, 64)
    diag_scale_kernel<<<grid, THREADS, 0, stream>>>(xr, xi, br, bi, out, hw, n_half);
}